// VanillaGaussian_79731772883646
// MI455X (gfx1250) — compile-verified
//
#include <hip/hip_runtime.h>

typedef float f4 __attribute__((ext_vector_type(4)));
typedef float f2 __attribute__((ext_vector_type(2)));

// 3D Gaussian -> 2D projection. Memory-bound streaming kernel:
// 40B in + 24B out per gaussian, ~200 flops. One gaussian per lane.
__global__ __launch_bounds__(256) void gsplat_project_kernel(
    const float* __restrict__ means,    // [N,3]
    const float* __restrict__ scales,   // [N,3]
    const f4*    __restrict__ quats,    // [N,4]
    const float* __restrict__ camR,     // [3,3] row-major
    const float* __restrict__ camT,     // [3]
    const float* __restrict__ fxp, const float* __restrict__ fyp,
    const float* __restrict__ cxp, const float* __restrict__ cyp,
    f2* __restrict__ center2d,          // [N,2]
    f4* __restrict__ sigma2d,           // [N,2,2] row-major
    int n)
{
    // Uniform camera params: uniform addresses -> scalar loads, L0-resident.
    const float R00 = camR[0], R01 = camR[1], R02 = camR[2];
    const float R10 = camR[3], R11 = camR[4], R12 = camR[5];
    const float R20 = camR[6], R21 = camR[7], R22 = camR[8];
    const float t0 = camT[0], t1 = camT[1], t2 = camT[2];
    const float fx = *fxp, fy = *fyp, cx = *cxp, cy = *cyp;

    int i = blockIdx.x * blockDim.x + threadIdx.x;
    const int stride = gridDim.x * blockDim.x;

    for (; i < n; i += stride) {
        // gfx1250 global_prefetch_b8 for next grid-stride trip.
        // Speculative: OOB translation failures are silently dropped (ISA 7.3).
        {
            const size_t j = (size_t)i + (size_t)stride;
            __builtin_prefetch(&means[3 * j], 0, 0);
            __builtin_prefetch(&scales[3 * j], 0, 0);
            __builtin_prefetch(&quats[j], 0, 0);
        }

        const size_t i3 = 3 * (size_t)i;

        // Streamed-once inputs: non-temporal loads (TH=NT) keep L2 clean.
        const f4 q = __builtin_nontemporal_load(&quats[i]);
        const float mx = __builtin_nontemporal_load(&means[i3 + 0]);
        const float my = __builtin_nontemporal_load(&means[i3 + 1]);
        const float mz = __builtin_nontemporal_load(&means[i3 + 2]);
        const float sx = __builtin_nontemporal_load(&scales[i3 + 0]);
        const float sy = __builtin_nontemporal_load(&scales[i3 + 1]);
        const float sz = __builtin_nontemporal_load(&scales[i3 + 2]);

        // Normalize quaternion (x, y, z, w convention).
        const float qinv = 1.0f / sqrtf(q.x * q.x + q.y * q.y + q.z * q.z + q.w * q.w);
        const float x = q.x * qinv, y = q.y * qinv, z = q.z * qinv, w = q.w * qinv;

        // Rotation matrix from quaternion.
        const float r00 = 1.0f - 2.0f * (y * y + z * z);
        const float r01 = 2.0f * (x * y - w * z);
        const float r02 = 2.0f * (x * z + w * y);
        const float r10 = 2.0f * (x * y + w * z);
        const float r11 = 1.0f - 2.0f * (x * x + z * z);
        const float r12 = 2.0f * (y * z - w * x);
        const float r20 = 2.0f * (x * z - w * y);
        const float r21 = 2.0f * (y * z + w * x);
        const float r22 = 1.0f - 2.0f * (x * x + y * y);

        // M = R * diag(exp(scales))  (column scaling)
        const float s0 = expf(sx), s1 = expf(sy), s2 = expf(sz);
        const float m00 = r00 * s0, m01 = r01 * s1, m02 = r02 * s2;
        const float m10 = r10 * s0, m11 = r11 * s1, m12 = r12 * s2;
        const float m20 = r20 * s0, m21 = r21 * s1, m22 = r22 * s2;

        // Sigma_world = M * M^T (symmetric, 6 unique entries)
        const float w00 = m00 * m00 + m01 * m01 + m02 * m02;
        const float w01 = m00 * m10 + m01 * m11 + m02 * m12;
        const float w02 = m00 * m20 + m01 * m21 + m02 * m22;
        const float w11 = m10 * m10 + m11 * m11 + m12 * m12;
        const float w12 = m10 * m20 + m11 * m21 + m12 * m22;
        const float w22 = m20 * m20 + m21 * m21 + m22 * m22;

        // Camera-space mean: mc = camR @ mean + t
        const float xc = R00 * mx + R01 * my + R02 * mz + t0;
        const float yc = R10 * mx + R11 * my + R12 * mz + t1;
        const float zc = R20 * mx + R21 * my + R22 * mz + t2;

        // Sigma_camera = camR @ W @ camR^T  (A = camR @ W, W symmetric)
        const float a00 = R00 * w00 + R01 * w01 + R02 * w02;
        const float a01 = R00 * w01 + R01 * w11 + R02 * w12;
        const float a02 = R00 * w02 + R01 * w12 + R02 * w22;
        const float a10 = R10 * w00 + R11 * w01 + R12 * w02;
        const float a11 = R10 * w01 + R11 * w11 + R12 * w12;
        const float a12 = R10 * w02 + R11 * w12 + R12 * w22;
        const float a20 = R20 * w00 + R21 * w01 + R22 * w02;
        const float a21 = R20 * w01 + R21 * w11 + R22 * w12;
        const float a22 = R20 * w02 + R21 * w12 + R22 * w22;

        const float c00 = a00 * R00 + a01 * R01 + a02 * R02;
        const float c01 = a00 * R10 + a01 * R11 + a02 * R12;
        const float c02 = a00 * R20 + a01 * R21 + a02 * R22;
        const float c11 = a10 * R10 + a11 * R11 + a12 * R12;
        const float c12 = a10 * R20 + a11 * R21 + a12 * R22;
        const float c22 = a20 * R20 + a21 * R21 + a22 * R22;

        // Perspective projection + Jacobian.
        const float rz = 1.0f / zc;
        const float rz2 = rz * rz;
        const float u = fx * xc * rz + cx;
        const float v = fy * yc * rz + cy;

        const float j00 = fx * rz;
        const float j02 = -fx * xc * rz2;
        const float j11 = fy * rz;
        const float j12 = -fy * yc * rz2;

        // B = J @ Sigma_camera  (J sparse: row0 = [j00, 0, j02], row1 = [0, j11, j12])
        const float b00 = j00 * c00 + j02 * c02;
        const float b01 = j00 * c01 + j02 * c12;
        const float b02 = j00 * c02 + j02 * c22;
        const float b10 = j11 * c01 + j12 * c02;
        const float b11 = j11 * c11 + j12 * c12;
        const float b12 = j11 * c12 + j12 * c22;

        // Sigma_2d = B @ J^T (write full 2x2 like the reference einsum)
        f4 s2d;
        s2d.x = b00 * j00 + b02 * j02;  // [0,0]
        s2d.y = b01 * j11 + b02 * j12;  // [0,1]
        s2d.z = b10 * j00 + b12 * j02;  // [1,0]
        s2d.w = b11 * j11 + b12 * j12;  // [1,1]

        f2 ctr;
        ctr.x = u;
        ctr.y = v;

        // Write-once outputs: non-temporal packed stores (b64 + b128, TH=NT).
        __builtin_nontemporal_store(ctr, &center2d[i]);
        __builtin_nontemporal_store(s2d, &sigma2d[i]);
    }
}

extern "C" void kernel_launch(void* const* d_in, const int* in_sizes, int n_in,
                              void* d_out, int out_size, void* d_ws, size_t ws_size,
                              hipStream_t stream) {
    // Input order from setup_inputs():
    // 0 means[N,3], 1 scales[N,3], 2 quats[N,4], 3 opacities[N,1] (UNUSED),
    // 4 cam_R[3,3], 5 cam_t[3], 6 fx, 7 fy, 8 cx, 9 cy
    const float* means  = (const float*)d_in[0];
    const float* scales = (const float*)d_in[1];
    const f4*    quats  = (const f4*)d_in[2];
    const float* camR   = (const float*)d_in[4];
    const float* camT   = (const float*)d_in[5];
    const float* fx     = (const float*)d_in[6];
    const float* fy     = (const float*)d_in[7];
    const float* cx     = (const float*)d_in[8];
    const float* cy     = (const float*)d_in[9];

    const int n = in_sizes[0] / 3;

    // d_out = [center_2d (N*2 floats) | sigma_2d (N*4 floats)]
    float* out = (float*)d_out;
    f2* center = (f2*)out;
    f4* sigma  = (f4*)(out + (size_t)2 * (size_t)n);

    const int threads = 256;  // 8 wave32s per block
    int blocksNeeded = (n + threads - 1) / threads;
    int blocks = blocksNeeded < 4096 ? blocksNeeded : 4096;  // ~4 grid-stride trips
    if (blocks < 1) blocks = 1;

    gsplat_project_kernel<<<blocks, threads, 0, stream>>>(
        means, scales, quats, camR, camT, fx, fy, cx, cy, center, sigma, n);
}